// FlowProjector_6468220748617
// MI455X (gfx1250) — compile-verified
//
#include <hip/hip_runtime.h>
#include <math.h>

// Sizes fixed by the reference problem.
#define S_LEN 511
#define D_IN  512
#define H1    128
#define H2    32
#define NT    32

typedef __attribute__((ext_vector_type(2))) float v2f;
typedef __attribute__((ext_vector_type(8))) float v8f;

__device__ __forceinline__ float nac_w(float w_hat, float m_hat) {
    // tanh(W_hat) * sigmoid(M_hat), full-precision f32 like the reference
    return tanhf(w_hat) * (1.0f / (1.0f + expf(-m_hat)));
}

// ---------------------------------------------------------------------------
// Stage 1: one 256-thread workgroup computes coef[0..31]=a3, coef[32..63]=b3
//   a = W1eff[:,0]            (128)
//   b = W1eff[:,1:] @ X       (128)
//   C = W3eff @ W2eff         (32x128)  -- via V_WMMA_F32_16X16X4_F32
//   a3 = C@a, b3 = C@b        (32 each)
// ---------------------------------------------------------------------------
__global__ __launch_bounds__(256)
void stage1_coeffs(const float* __restrict__ X,
                   const float* __restrict__ Wh1, const float* __restrict__ Mh1,
                   const float* __restrict__ Wh2, const float* __restrict__ Mh2,
                   const float* __restrict__ Wh3, const float* __restrict__ Mh3,
                   float* __restrict__ coef) {
    __shared__ float a_v[H1];
    __shared__ float b_v[H1];
    __shared__ float pb[256];
    __shared__ float W2s[H2][H1];   // 32x128 effective W2
    __shared__ float W3s[H2][H2];   // 32x32  effective W3
    __shared__ float Cs[H2][H1];    // 32x128 composite W3@W2

    const int tid = threadIdx.x;

    // ---- phase 1: a (first column) and b = W1eff[:,1:] @ X -----------------
    {
        const int m    = tid >> 1;      // row 0..127
        const int half = tid & 1;       // column half
        const float* wr = Wh1 + (size_t)m * D_IN;
        const float* mr = Mh1 + (size_t)m * D_IN;
        const int c0 = half * 256;
        float s = 0.0f;
        #pragma unroll 4
        for (int i = 0; i < 256; ++i) {
            const int c = c0 + i;
            const float w = nac_w(wr[c], mr[c]);
            if (c == 0) {
                a_v[m] = w;             // first column -> 'a'
            } else {
                s = fmaf(w, X[c - 1], s);
            }
        }
        pb[tid] = s;
    }

    // ---- phase 1b: effective W2, W3 into LDS ------------------------------
    for (int i = tid; i < H2 * H1; i += 256)
        (&W2s[0][0])[i] = nac_w(Wh2[i], Mh2[i]);
    for (int i = tid; i < H2 * H2; i += 256)
        (&W3s[0][0])[i] = nac_w(Wh3[i], Mh3[i]);
    __syncthreads();

    if (tid < H1) b_v[tid] = pb[2 * tid] + pb[2 * tid + 1];
    __syncthreads();

    // ---- phase 2: C = W3eff @ W2eff via f32 WMMA --------------------------
    // 8 waves (wave32), tile grid 2(M) x 4(N) of 16x16 tiles, K-loop 8 x K=4.
    {
        const int lane = tid & 31;
        const int wv   = tid >> 5;          // 0..7
        const int tm   = wv & 1;            // M-tile 0..1
        const int tn   = wv >> 1;           // N-tile 0..3
        const int lrow = lane & 15;
        const int lhi  = lane >> 4;         // 0: lanes 0-15, 1: lanes 16-31

        v8f acc = {};
        #pragma unroll
        for (int kk = 0; kk < 8; ++kk) {
            const int k = kk * 4 + lhi * 2; // upper lanes hold K+2 (ISA layout)
            v2f A, B;
            // A: 16x4 tile of W3eff, rows tm*16+lrow, VGPR0=K, VGPR1=K+1
            A.x = W3s[tm * 16 + lrow][k];
            A.y = W3s[tm * 16 + lrow][k + 1];
            // B: 4x16 tile of W2eff, cols tn*16+lrow, row striped across lanes
            B.x = W2s[k][tn * 16 + lrow];
            B.y = W2s[k + 1][tn * 16 + lrow];
            acc = __builtin_amdgcn_wmma_f32_16x16x4_f32(
                false, A, false, B, (short)0, acc, false, false);
        }
        // D layout: VGPR r -> rows tm*16 + r (lanes 0-15) and +8 (lanes 16-31)
        #pragma unroll
        for (int r = 0; r < 8; ++r)
            Cs[tm * 16 + r + lhi * 8][tn * 16 + lrow] = acc[r];
    }
    __syncthreads();

    // ---- phase 3: a3 = C@a, b3 = C@b --------------------------------------
    if (tid < 64) {
        const int j = tid & 31;
        const float* v = (tid < 32) ? a_v : b_v;
        float s = 0.0f;
        #pragma unroll 8
        for (int m = 0; m < H1; ++m)
            s = fmaf(Cs[j][m], v[m], s);
        coef[tid] = s;   // [0..31]=a3, [32..63]=b3
    }
}

// ---------------------------------------------------------------------------
// Stage 2: out[t*32 + j] = (t/T)*a3[j] + b3[j]
// One float4 per thread, consecutive float4 indices -> coalesced b128 stores.
// 32 MB store stream: HBM-bound (~1.4 us at 23.3 TB/s).
// ---------------------------------------------------------------------------
__global__ __launch_bounds__(256)
void stage2_fill(const float* __restrict__ coef, float* __restrict__ out,
                 int T, float inv_n) {
    __shared__ float4 c[16];                 // c[0..7]=a3, c[8..15]=b3
    const int tid = threadIdx.x;
    if (tid < 16) c[tid] = ((const float4*)coef)[tid];
    __syncthreads();

    const long long idx = (long long)blockIdx.x * 256 + tid;  // float4 index
    if (idx >= (long long)T * 8) return;
    const int t = (int)(idx >> 3);
    const int q = (int)(idx & 7);
    const float tv = (float)t * inv_n;       // exact: inv_n is 2^-k
    const float4 A = c[q];
    const float4 B = c[8 + q];
    float4 r;
    r.x = fmaf(A.x, tv, B.x);
    r.y = fmaf(A.y, tv, B.y);
    r.z = fmaf(A.z, tv, B.z);
    r.w = fmaf(A.w, tv, B.w);
    ((float4*)out)[idx] = r;
}

extern "C" void kernel_launch(void* const* d_in, const int* in_sizes, int n_in,
                              void* d_out, int out_size, void* d_ws, size_t ws_size,
                              hipStream_t stream) {
    const float* X   = (const float*)d_in[0];
    const float* Wh1 = (const float*)d_in[1];
    const float* Mh1 = (const float*)d_in[2];
    const float* Wh2 = (const float*)d_in[3];
    const float* Mh2 = (const float*)d_in[4];
    const float* Wh3 = (const float*)d_in[5];
    const float* Mh3 = (const float*)d_in[6];
    // n_steps lives on device; host derives T from out_size = T * 32.
    const int T = out_size / NT;

    float* coef = (float*)d_ws;              // 64 floats of scratch
    float* out  = (float*)d_out;

    stage1_coeffs<<<1, 256, 0, stream>>>(X, Wh1, Mh1, Wh2, Mh2, Wh3, Mh3, coef);

    const long long nvec = (long long)T * 8; // float4 count
    const int blocks = (int)((nvec + 255) / 256);
    stage2_fill<<<blocks, 256, 0, stream>>>(coef, out, T, 1.0f / (float)T);
}